// GraphProjection_8203387535722
// MI455X (gfx1250) — compile-verified
//
#include <hip/hip_runtime.h>

// ---------------------------------------------------------------------------
// GraphProjection for MI455X (gfx1250, wave32).
// Store-bandwidth-bound: ~505 MB of output vs ~1.5 MB of (L2-resident) reads.
// CDNA5 paths: TDM tensor_load_to_lds (feat4 -> LDS), s_wait_tensorcnt,
// ds_load_b128 gathers from LDS, non-temporal streaming stores for output.
// ---------------------------------------------------------------------------

typedef __attribute__((ext_vector_type(4))) unsigned int u32x4;
typedef __attribute__((ext_vector_type(8))) int          i32x8;
typedef __attribute__((ext_vector_type(4))) int          i32x4;
typedef __attribute__((ext_vector_type(4))) float        f32x4;
typedef __attribute__((ext_vector_type(2))) float        f32x2;

#define DEV_INLINE __device__ __forceinline__

// Shared bilinear setup: reproduces the reference's floor/ceil weights exactly
// (integer coords -> all four weights zero, matching the JAX semantics).
struct BilinSetup {
  int i11, i21, i12, i22;      // element offsets of the 4 corner rows
  float w11, w21, w12, w22;
};

template <int S, int C>
DEV_INLINE BilinSetup bilin_setup(float h, float w) {
  constexpr float inv = (float)S / 224.0f;  // 0.25/0.125/0.0625/0.03125: exact
  const float x = h * inv, y = w * inv;
  const float x1 = floorf(x), x2 = ceilf(x);
  const float y1 = floorf(y), y2 = ceilf(y);
  const int xi1 = (int)x1, xi2 = (int)x2, yi1 = (int)y1, yi2 = (int)y2;
  const float wx1 = x2 - x, wx2 = x - x1;
  const float wy1 = y2 - y, wy2 = y - y1;
  BilinSetup s;
  s.w11 = wx1 * wy1; s.w21 = wx2 * wy1; s.w12 = wx1 * wy2; s.w22 = wx2 * wy2;
  s.i11 = (xi1 * S + yi1) * C; s.i21 = (xi2 * S + yi1) * C;
  s.i12 = (xi1 * S + yi2) * C; s.i22 = (xi2 * S + yi2) * C;
  return s;
}

// Global-memory level. VW = floats per lane (2 for C=64 so all 32 lanes work).
template <int S, int C, int VW>
DEV_INLINE void project_level(const float* __restrict__ feat, float h, float w,
                              int lane, float* __restrict__ orow) {
  const BilinSetup s = bilin_setup<S, C>(h, w);
  for (int c = lane * VW; c < C; c += 32 * VW) {
    if constexpr (VW == 4) {
      const f32x4 a = *(const f32x4*)(feat + s.i11 + c);
      const f32x4 b = *(const f32x4*)(feat + s.i21 + c);
      const f32x4 d = *(const f32x4*)(feat + s.i12 + c);
      const f32x4 e = *(const f32x4*)(feat + s.i22 + c);
      __builtin_nontemporal_store(s.w11*a.x + s.w21*b.x + s.w12*d.x + s.w22*e.x, orow + c + 0);
      __builtin_nontemporal_store(s.w11*a.y + s.w21*b.y + s.w12*d.y + s.w22*e.y, orow + c + 1);
      __builtin_nontemporal_store(s.w11*a.z + s.w21*b.z + s.w12*d.z + s.w22*e.z, orow + c + 2);
      __builtin_nontemporal_store(s.w11*a.w + s.w21*b.w + s.w12*d.w + s.w22*e.w, orow + c + 3);
    } else {
      const f32x2 a = *(const f32x2*)(feat + s.i11 + c);
      const f32x2 b = *(const f32x2*)(feat + s.i21 + c);
      const f32x2 d = *(const f32x2*)(feat + s.i12 + c);
      const f32x2 e = *(const f32x2*)(feat + s.i22 + c);
      __builtin_nontemporal_store(s.w11*a.x + s.w21*b.x + s.w12*d.x + s.w22*e.x, orow + c + 0);
      __builtin_nontemporal_store(s.w11*a.y + s.w21*b.y + s.w12*d.y + s.w22*e.y, orow + c + 1);
    }
  }
}

// LDS-resident level (feat4): explicit ds_load_b128 gathers.
// lds_base is the LDS byte offset of the staged array (low 32 bits of its
// flat address, per the ISA's LDS aperture address calculation).
template <int S, int C>
DEV_INLINE void project_level_lds(unsigned lds_base, float h, float w,
                                  int lane, float* __restrict__ orow) {
  const BilinSetup s = bilin_setup<S, C>(h, w);
  const unsigned b11 = lds_base + (unsigned)s.i11 * 4u;
  const unsigned b21 = lds_base + (unsigned)s.i21 * 4u;
  const unsigned b12 = lds_base + (unsigned)s.i12 * 4u;
  const unsigned b22 = lds_base + (unsigned)s.i22 * 4u;
  for (int c = lane * 4; c < C; c += 128) {
    const unsigned cb = (unsigned)c * 4u;
    f32x4 a, b, d, e;
    asm volatile("ds_load_b128 %0, %4\n\t"
                 "ds_load_b128 %1, %5\n\t"
                 "ds_load_b128 %2, %6\n\t"
                 "ds_load_b128 %3, %7\n\t"
                 "s_wait_dscnt 0x0"
                 : "=&v"(a), "=&v"(b), "=&v"(d), "=&v"(e)
                 : "v"(b11 + cb), "v"(b21 + cb), "v"(b12 + cb), "v"(b22 + cb));
    __builtin_nontemporal_store(s.w11*a.x + s.w21*b.x + s.w12*d.x + s.w22*e.x, orow + c + 0);
    __builtin_nontemporal_store(s.w11*a.y + s.w21*b.y + s.w12*d.y + s.w22*e.y, orow + c + 1);
    __builtin_nontemporal_store(s.w11*a.z + s.w21*b.z + s.w12*d.z + s.w22*e.z, orow + c + 2);
    __builtin_nontemporal_store(s.w11*a.w + s.w21*b.w + s.w12*d.w + s.w22*e.w, orow + c + 3);
  }
}

__global__ __launch_bounds__(256) void GraphProjection_kernel(
    const float* __restrict__ coord,
    const float* __restrict__ feat1, const float* __restrict__ feat2,
    const float* __restrict__ feat3, const float* __restrict__ feat4,
    float* __restrict__ out, int n) {
  // feat4 = 7*7*512 floats = 100,352 bytes in LDS.
  __shared__ float sfeat4[7 * 7 * 512];
  // LDS byte offset of sfeat4 = low 32 bits of its flat address (ISA §10.2).
  const unsigned lds_base = (unsigned)(unsigned long long)(const void*)sfeat4;

  // --- Stage feat4 into LDS with the Tensor Data Mover (one wave issues) ---
  if (threadIdx.x < 32) {
    const unsigned long long ga = (unsigned long long)feat4;
    // D# group 0: count=1 (valid), lds_addr, global_addr, type=2 ("image")
    u32x4 g0;
    g0.x = 1u;
    g0.y = lds_base;
    g0.z = (unsigned)(ga & 0xffffffffull);
    g0.w = (unsigned)((ga >> 32) & 0x01ffffffull) | 0x80000000u;  // type=2 at [127:126]
    // D# group 1: data_size=4B, 1-D tile covering all 25088 elements.
    i32x8 g1;
    g1[0] = 0x00020000;                 // wg_mask=0, data_size=2 (4B), no pad/iterate
    g1[1] = (int)(25088u << 16);        // tensor_dim0[15:0]=25088
    g1[2] = 0x00010000;                 // tensor_dim0[31:16]=0, tensor_dim1=1
    g1[3] = (int)(25088u << 16);        // tile_dim0=25088
    g1[4] = 0;                          // tile_dim1=0 (unused), tile_dim2=0
    g1[5] = 25088;                      // tensor_dim0_stride
    g1[6] = 0;
    g1[7] = 0;
    i32x4 g2; g2[0] = 1; g2[1] = 1; g2[2] = 0; g2[3] = 0;  // dim2/dim3 = 1 (unused)
    i32x4 g3; g3[0] = 0; g3[1] = 0; g3[2] = 0; g3[3] = 0;
    i32x8 g4; g4[0] = 0; g4[1] = 0; g4[2] = 0; g4[3] = 0;
              g4[4] = 0; g4[5] = 0; g4[6] = 0; g4[7] = 0;
    __builtin_amdgcn_tensor_load_to_lds(g0, g1, g2, g3, g4, 0);
    __builtin_amdgcn_s_wait_tensorcnt(0);
  }
  __syncthreads();

  const int lane   = threadIdx.x & 31;
  const int wave   = (int)((blockIdx.x * blockDim.x + threadIdx.x) >> 5);
  const int nwaves = (int)((gridDim.x * blockDim.x) >> 5);

  // Strided point assignment: concurrent waves write adjacent 3852B rows.
  for (int p = wave; p < n; p += nwaves) {
    const float X = coord[3 * p + 0];
    const float Y = coord[3 * p + 1];
    const float Z = coord[3 * p + 2];
    float* orow = out + (long long)p * 963;
    if (lane < 3) {
      const float c3[3] = {X, Y, Z};
      __builtin_nontemporal_store(c3[lane], orow + lane);
    }
    const float nz = -Z;                 // Z <= -1, so nz >= 1 (no div-by-0)
    float h = 248.0f * ((-Y) / nz) + 112.0f;
    float w = 248.0f * (X / nz) + 112.0f;
    h = fminf(fmaxf(h, 0.0f), 223.0f);
    w = fminf(fmaxf(w, 0.0f), 223.0f);

    project_level<56,  64, 2>(feat1, h, w, lane, orow + 3);    // 32 lanes x 2
    project_level<28, 128, 4>(feat2, h, w, lane, orow + 67);
    project_level<14, 256, 4>(feat3, h, w, lane, orow + 195);
    project_level_lds<7, 512>(lds_base, h, w, lane, orow + 451);  // DS path
  }
}

extern "C" void kernel_launch(void* const* d_in, const int* in_sizes, int n_in,
                              void* d_out, int out_size, void* d_ws, size_t ws_size,
                              hipStream_t stream) {
  const float* coord = (const float*)d_in[0];
  const float* feat1 = (const float*)d_in[1];
  const float* feat2 = (const float*)d_in[2];
  const float* feat3 = (const float*)d_in[3];
  const float* feat4 = (const float*)d_in[4];
  float* out = (float*)d_out;
  const int n = in_sizes[0] / 3;  // 131072 points

  // 512 blocks x 256 threads = 4096 waves; 32 points/wave; ~100KB LDS/block
  // -> up to 3 blocks per WGP (24 waves) for latency hiding on a
  // store-bandwidth-bound kernel.
  dim3 grid(512), block(256);
  GraphProjection_kernel<<<grid, block, 0, stream>>>(coord, feat1, feat2, feat3,
                                                     feat4, out, n);
}